// tiling_31464930411246
// MI455X (gfx1250) — compile-verified
//
#include <hip/hip_runtime.h>

// Depth-to-space (pixel shuffle), r=4:
//   y[b, od, 4h+rh, 4w+rw] = x[b, od*16 + rh*4 + rw, h, w]
// x: (8, 256, 128, 128) f32,  y: (8, 16, 512, 512) f32.
//
// Pure data movement: 268 MB total traffic, ~11.5 us floor at 23.3 TB/s.
// Each lane: 4x coalesced global_load_b128 (one per rw-channel, same address
// VGPR pair + 64KB immediate offsets), free 4x4 register transpose, then
// 4x coalesced global_store_b128 into one contiguous 64B output segment.
// Non-temporal hints on both sides: 256 MB stream > 192 MB L2.

typedef float v4f __attribute__((ext_vector_type(4)));

__global__ __launch_bounds__(256, 8)
void d2s_r4_kernel(const float* __restrict__ x, float* __restrict__ y,
                   unsigned n_seg) {
    constexpr unsigned W = 128, H = 128;         // input spatial
    constexpr unsigned R = 4, OD = 16;           // tile, out channels
    constexpr unsigned WO = W * R;               // 512 (output width == output height)
    constexpr unsigned CH = W * H;               // 16384 floats between channels (64 KB)

    unsigned seg = blockIdx.x * 256u + threadIdx.x;   // one 16-float output segment
    if (seg >= n_seg) return;

    // Decompose: 32 segments per output row; rows ordered (b, od, h_out).
    unsigned sw    = seg & 31u;                 // segment within output row (== lane)
    unsigned row   = seg >> 5;
    unsigned h_out = row & (WO - 1u);           // 0..511
    unsigned od    = (row >> 9) & (OD - 1u);
    unsigned b     = row >> 13;
    unsigned rh    = h_out & (R - 1u);
    unsigned h     = h_out >> 2;

    // Input: channel block c0 = od*16 + rh*4, float4 along w at w = 4*sw.
    size_t in_base = ((((size_t)b * 256u + (od * 16u + rh * 4u)) * (size_t)H + h)
                      * (size_t)W) + sw * 4u;
    v4f a0 = __builtin_nontemporal_load((const v4f*)(x + in_base));           // rw=0
    v4f a1 = __builtin_nontemporal_load((const v4f*)(x + in_base +     CH));  // rw=1
    v4f a2 = __builtin_nontemporal_load((const v4f*)(x + in_base + 2 * CH));  // rw=2
    v4f a3 = __builtin_nontemporal_load((const v4f*)(x + in_base + 3 * CH));  // rw=3

    // 4x4 register transpose: out[j] = a_{j&3}[j>>2], j = 0..15.
    v4f o0 = {a0.x, a1.x, a2.x, a3.x};
    v4f o1 = {a0.y, a1.y, a2.y, a3.y};
    v4f o2 = {a0.z, a1.z, a2.z, a3.z};
    v4f o3 = {a0.w, a1.w, a2.w, a3.w};

    // Output: contiguous 64 B segment at (b, od, h_out, 16*sw).
    size_t out_base = (((size_t)(b * OD + od) * (size_t)WO + h_out)
                       * (size_t)WO) + sw * 16u;
    __builtin_nontemporal_store(o0, (v4f*)(y + out_base +  0));
    __builtin_nontemporal_store(o1, (v4f*)(y + out_base +  4));
    __builtin_nontemporal_store(o2, (v4f*)(y + out_base +  8));
    __builtin_nontemporal_store(o3, (v4f*)(y + out_base + 12));
}

extern "C" void kernel_launch(void* const* d_in, const int* in_sizes, int n_in,
                              void* d_out, int out_size, void* d_ws, size_t ws_size,
                              hipStream_t stream) {
    const float* x = (const float*)d_in[0];
    float*       y = (float*)d_out;

    // 16 output floats per thread; out_size = 8*16*512*512 = 33,554,432
    unsigned n_seg  = (unsigned)(out_size / 16);      // 2,097,152
    unsigned blocks = (n_seg + 255u) / 256u;          // 8192
    d2s_r4_kernel<<<blocks, 256, 0, stream>>>(x, y, n_seg);
}